// Con_sp_4131758539348
// MI455X (gfx1250) — compile-verified
//
#include <hip/hip_runtime.h>
#include <hip/hip_bf16.h>
#include <stdint.h>

// ---------------------------------------------------------------------------
// LeNet C3 sparse conv == one dense 5x5x6x16 VALID conv.
// Implicit GEMM on the CDNA5 matrix pipe with V_WMMA_F32_16X16X4_F32:
//   M = output pixels, N = 16 output channels, K = 5*5*6 = 150 (padded to 152)
// Input halo + weight fragments staged to LDS with the Tensor Data Mover.
// ---------------------------------------------------------------------------

#define BATCH 32
#define H     512
#define W     512
#define CIN   6
#define COUT  16
#define FS    5
#define OH    508
#define OW    508

#define TILE_X   128                 // output pixels per block (8 waves x 16)
#define IN_COLS  (TILE_X + FS - 1)   // 132 input columns staged
#define ROW_F    (IN_COLS * CIN)     // 792 floats per staged input row
#define KPAD     152                 // 150 rounded up to multiple of 4
#define NCHUNK   (KPAD / 4)          // 38 WMMA steps
#define BFRAG_F  (NCHUNK * 2 * 32)   // 2432 floats of pre-laid-out B fragments

typedef __attribute__((ext_vector_type(2))) float v2f;
typedef __attribute__((ext_vector_type(8))) float v8f;
typedef __attribute__((ext_vector_type(4))) unsigned int u32x4;
typedef __attribute__((ext_vector_type(8))) int i32x8;
typedef __attribute__((ext_vector_type(4))) int i32x4;

#if __has_builtin(__builtin_amdgcn_tensor_load_to_lds)
#define USE_TDM 1
#else
#define USE_TDM 0
#endif

// k = (fy*5 + fx)*6 + c  ->  LDS offset within staged input tile
__host__ __device__ constexpr int koff(int k) {
    return (k / 30) * ROW_F + ((k / 6) % 5) * CIN + (k % 6);
}

// Low 32 bits of a generic LDS pointer = LDS byte offset (flat-LDS mapping).
__device__ inline uint32_t lds_off(const void* p) {
    return (uint32_t)(uintptr_t)p;
}

#if USE_TDM
// Issue one TDM 2D tile load: tile_d1 rows of tile_d0 elements (4B each),
// global row stride stride_d0 elements. Tensor dims set huge: the window is
// in-bounds by construction, so no OOB zero-fill path is exercised.
__device__ inline void tdm_load_2d(uint32_t lds_addr, const void* gptr,
                                   uint32_t tile_d0, uint32_t tile_d1,
                                   uint32_t stride_d0) {
    const uint64_t ga  = (uint64_t)(uintptr_t)gptr;
    const uint32_t td0 = 1u << 30, td1 = 1u << 30;
    u32x4 g0;
    g0[0] = 1u;                                          // count=1, user D#
    g0[1] = lds_addr;                                    // lds_addr (bytes)
    g0[2] = (uint32_t)ga;                                // global_addr[31:0]
    g0[3] = ((uint32_t)(ga >> 32) & 0x01FFFFFFu)         // global_addr[56:32]
            | (2u << 30);                                // type=2 ("image")
    i32x8 g1;
    g1[0] = (int)(2u << 16);                             // data_size=4B; mask=0
    g1[1] = (int)((td0 & 0xFFFFu) << 16);                // tensor_dim0[15:0]
    g1[2] = (int)(((td0 >> 16) & 0xFFFFu) |
                  ((td1 & 0xFFFFu) << 16));              // td0 hi | td1 lo
    g1[3] = (int)(((td1 >> 16) & 0xFFFFu) |
                  ((tile_d0 & 0xFFFFu) << 16));          // td1 hi | tile_dim0
    g1[4] = (int)(tile_d1 & 0xFFFFu);                    // tile_dim1; dim2=0
    g1[5] = (int)stride_d0;                              // dim0_stride low32
    g1[6] = 0;                                           // stride hi | d1str lo
    g1[7] = 0;
    i32x4 z4 = {0, 0, 0, 0};                             // groups 2/3 unused
    i32x8 z8 = {0, 0, 0, 0, 0, 0, 0, 0};                 // trailing group unused
    __builtin_amdgcn_tensor_load_to_lds(g0, g1, z4, z4, z8, 0);
}
#endif

// ---------------------------------------------------------------------------
// Prep: assemble the dense (sparse-connection) kernel directly into the
// B-fragment lane layout for V_WMMA_F32_16X16X4_F32, K-padded with zeros.
// bfrag[kc][v][lane]: v0 holds K-rows 4kc (lanes 0-15) / 4kc+2 (lanes 16-31),
//                     v1 holds K-rows 4kc+1 / 4kc+3. N = lane & 15.
// ---------------------------------------------------------------------------
__global__ void c3_prep_bfrag(const float* __restrict__ w3,
                              const float* __restrict__ w4,
                              const float* __restrict__ w44,
                              const float* __restrict__ w6,
                              float* __restrict__ bfrag) {
    int idx = blockIdx.x * blockDim.x + threadIdx.x;
    if (idx >= KPAD * COUT) return;
    int k = idx / COUT;     // 0..151
    int n = idx % COUT;     // output channel
    float v = 0.0f;
    if (k < 150) {
        int fy = k / 30, rem = k % 30, fx = rem / 6, c = rem % 6;
        int pos = fy * FS + fx;                       // 0..24
        if (n < 6) {                                  // 3 consecutive inputs
            for (int m = 0; m < 3; ++m)
                if (((n + m) % 6) == c) v = w3[(pos * 3 + m) * 6 + n];
        } else if (n < 12) {                          // 4 consecutive inputs
            int kk = n - 6;
            for (int m = 0; m < 4; ++m)
                if (((kk + m) % 6) == c) v = w4[(pos * 4 + m) * 6 + kk];
        } else if (n < 15) {                          // offsets 0,1,3,4
            int kk = n - 12;
            const int offs[4] = {0, 1, 3, 4};
            for (int m = 0; m < 4; ++m)
                if (((kk + offs[m]) % 6) == c) v = w44[(pos * 4 + m) * 3 + kk];
        } else {                                      // all 6 inputs
            v = w6[pos * 6 + c];
        }
    }
    int kc = k >> 2, r = k & 3;
    int vreg = r & 1, half = r >> 1;
    bfrag[(kc * 2 + vreg) * 32 + half * 16 + n] = v;
}

// ---------------------------------------------------------------------------
// Main conv: one block = (b, y, 128-wide x tile); one wave = 16 pixels x 16 ch.
// ---------------------------------------------------------------------------
__global__ __launch_bounds__(256) void c3_conv_wmma(
        const float* __restrict__ in,     // [32,512,512,6]
        const float* __restrict__ bfrag,  // [38][2][32]
        const float* __restrict__ bias,   // [16]
        float* __restrict__ out) {        // [32,508,508,16]
    __shared__ float lds_in[FS * ROW_F];   // 3960 floats (15.84 KB)
    __shared__ float lds_b[BFRAG_F];       // 2432 floats ( 9.73 KB)

    const int tid  = threadIdx.x;
    const int lane = tid & 31;
    const int wave = tid >> 5;

    const int x0 = blockIdx.x * TILE_X;    // 0,128,256,384
    const int y  = blockIdx.y;             // 0..507
    const int b  = blockIdx.z;             // 0..31

    // Shift the staged window left at the right edge so all global reads are
    // in-bounds; invalid pixels (x >= 508) are masked only at the store.
    int xs = x0;
    if (xs > W - IN_COLS) xs = W - IN_COLS;   // 380 for the last tile
    const int shift = x0 - xs;                // 0 or 4

    const float* inb = in + (((size_t)b * H + y) * W + xs) * CIN;

#if USE_TDM
    if (wave == 0) {
        // 5 rows x 792 floats, global row stride 3072 floats -> lds_in
        tdm_load_2d(lds_off(lds_in), inb, ROW_F, FS, W * CIN);
        // 2432 contiguous floats of B fragments -> lds_b
        tdm_load_2d(lds_off(lds_b), bfrag, BFRAG_F, 1, BFRAG_F);
        __builtin_amdgcn_s_wait_tensorcnt(0);
    }
#else
    // Fallback: contiguous coalesced copies (no per-element div/mod/clamp).
    for (int i = tid; i < BFRAG_F; i += 256) lds_b[i] = bfrag[i];
#pragma unroll
    for (int fy = 0; fy < FS; ++fy)
        for (int j = tid; j < ROW_F; j += 256)
            lds_in[fy * ROW_F + j] = inb[(size_t)fy * W * CIN + j];
#endif
    __syncthreads();

    const int  p     = lane & 15;           // pixel within wave tile / channel
    const bool hi    = lane >= 16;
    const int  wx    = wave * 16;           // wave's x offset in block tile
    const int  abase = (shift + wx + p) * CIN;  // per-lane A base in LDS

    v8f acc = {};

#pragma unroll
    for (int kc = 0; kc < NCHUNK; ++kc) {
        v2f a, bb;
        if (kc == NCHUNK - 1) {
            // k = 148,149 valid; 150,151 are zero padding (hi lanes)
            float a0 = lds_in[abase + koff(148)];
            float a1 = lds_in[abase + koff(149)];
            a.x = hi ? 0.0f : a0;
            a.y = hi ? 0.0f : a1;
        } else {
            int off0 = hi ? koff(kc * 4 + 2) : koff(kc * 4 + 0);
            int off1 = hi ? koff(kc * 4 + 3) : koff(kc * 4 + 1);
            a.x = lds_in[abase + off0];
            a.y = lds_in[abase + off1];
        }
        bb.x = lds_b[(kc * 2 + 0) * 32 + lane];
        bb.y = lds_b[(kc * 2 + 1) * 32 + lane];
        // D = A(16x4 f32) x B(4x16 f32) + C, full f32 accumulate
        acc = __builtin_amdgcn_wmma_f32_16x16x4_f32(
                /*neg_a=*/false, a, /*neg_b=*/false, bb,
                /*c_mod=*/(short)0, acc, /*reuse_a=*/false, /*reuse_b=*/false);
    }

    // --- epilogue: bias + masked store ---
    const float bv = bias[p];               // N = lane & 15
#pragma unroll
    for (int r = 0; r < 8; ++r) {
        int m = r + (hi ? 8 : 0);           // D layout: VGPR r -> M=r / r+8
        int x = x0 + wx + m;
        if (x < OW) {
            out[(((size_t)b * OH + y) * OW + x) * COUT + p] = acc[r] + bv;
        }
    }
}

extern "C" void kernel_launch(void* const* d_in, const int* in_sizes, int n_in,
                              void* d_out, int out_size, void* d_ws, size_t ws_size,
                              hipStream_t stream) {
    const float* in   = (const float*)d_in[0];  // inputs   [32,512,512,6]
    const float* w3   = (const float*)d_in[1];  // weights3 [5,5,3,6]
    const float* w4   = (const float*)d_in[2];  // weights4 [5,5,4,6]
    const float* w44  = (const float*)d_in[3];  // weights4_4 [5,5,4,3]
    const float* w6   = (const float*)d_in[4];  // weights6 [5,5,6,1]
    const float* bias = (const float*)d_in[5];  // bias1    [16]
    float* out   = (float*)d_out;
    float* bfrag = (float*)d_ws;                // 2432 floats = 9728 B

    c3_prep_bfrag<<<(KPAD * COUT + 255) / 256, 256, 0, stream>>>(w3, w4, w44, w6, bfrag);

    dim3 grid((OW + TILE_X - 1) / TILE_X, OH, BATCH);   // 4 x 508 x 32
    c3_conv_wmma<<<grid, 256, 0, stream>>>(in, bfrag, bias, out);
}